// Geo_Struct_59322088292888
// MI455X (gfx1250) — compile-verified
//
#include <hip/hip_runtime.h>
#include <hip/hip_bf16.h>

// Problem constants (match reference)
#define B_    16
#define H_    1024
#define W_    1024
#define K_    256
#define NSEG  (B_ * K_)          // 4096 instances total
#define TWO_PI 6.28318530717958647692f
#define IMAX_  0x7fffffff
#define IMIN_  (-0x7fffffff - 1)

typedef float v2f __attribute__((ext_vector_type(2)));
typedef float v8f __attribute__((ext_vector_type(8)));

// Workspace layout (ints): minx[NSEG] | miny[NSEG] | maxx[NSEG] | maxy[NSEG]
__global__ void geo_init_bbox(int* __restrict__ ws) {
    int i = blockIdx.x * blockDim.x + threadIdx.x;   // 0..16383
    ws[i] = (i < 2 * NSEG) ? IMAX_ : IMIN_;
}

// Each block: 16 rows of one image; thread = one 64-px row segment.
// Run-length compress ids -> few LDS atomics; merge touched ids to global.
__global__ void __launch_bounds__(256)
geo_scan(const int* __restrict__ imap, int* __restrict__ ws) {
    __shared__ int sminx[K_], sminy[K_], smaxx[K_], smaxy[K_];
    const int tid = threadIdx.x;
    sminx[tid] = IMAX_; sminy[tid] = IMAX_;
    smaxx[tid] = IMIN_; smaxy[tid] = IMIN_;
    __syncthreads();

    const int img     = blockIdx.x >> 6;          // 64 blocks per image
    const int rowBase = (blockIdx.x & 63) * 16;   // 16 rows per block
    const int rowIn   = tid >> 4;                 // 0..15
    const int segIn   = tid & 15;                 // 0..15
    const int y       = rowBase + rowIn;
    const int x0      = segIn * 64;

    const int4* p = reinterpret_cast<const int4*>(
        imap + ((size_t)img * H_ + y) * (size_t)W_ + x0);

    int cur = 0, rminx = 0, rmaxx = 0;
    #pragma unroll 4
    for (int i = 0; i < 16; ++i) {
        __builtin_prefetch(p + i + 4, 0, 0);   // global_prefetch_b8
        int4 v = p[i];
        const int xb = x0 + i * 4;
        int ids[4] = { v.x, v.y, v.z, v.w };
        #pragma unroll
        for (int j = 0; j < 4; ++j) {
            int id = ids[j];
            if (id == cur) {
                rmaxx = xb + j;
            } else {
                if (cur > 0 && cur <= K_) {
                    int li = cur - 1;
                    atomicMin(&sminx[li], rminx);
                    atomicMax(&smaxx[li], rmaxx);
                    atomicMin(&sminy[li], y);
                    atomicMax(&smaxy[li], y);
                }
                cur = id; rminx = xb + j; rmaxx = xb + j;
            }
        }
    }
    if (cur > 0 && cur <= K_) {
        int li = cur - 1;
        atomicMin(&sminx[li], rminx);
        atomicMax(&smaxx[li], rmaxx);
        atomicMin(&sminy[li], y);
        atomicMax(&smaxy[li], y);
    }
    __syncthreads();

    // thread tid merges instance tid of this image (only if touched)
    if (sminx[tid] != IMAX_) {
        int g = img * K_ + tid;
        atomicMin(&ws[g],            sminx[tid]);
        atomicMin(&ws[NSEG + g],     sminy[tid]);
        atomicMax(&ws[2 * NSEG + g], smaxx[tid]);
        atomicMax(&ws[3 * NSEG + g], smaxy[tid]);
    }
}

// coords(16 rows x K=4, K padded) x gauss-tile(4 x 16) via V_WMMA_F32_16X16X4_F32.
// Rows g in [0,8192): instance n = g>>1, point p = g&1 (tl / br corner).
// 8 WMMAs per wave cover all 128 gaussian dims; then 2*pi, sincos, +pe2/pe3.
__global__ void __launch_bounds__(256)
geo_embed(const int* __restrict__ ws,
          const float* __restrict__ gauss,
          const float* __restrict__ pe2,
          const float* __restrict__ pe3,
          float* __restrict__ out) {
    const int lane = threadIdx.x & 31;
    const int wave = threadIdx.x >> 5;
    const int rowBase = (blockIdx.x * 8 + wave) * 16;   // 8192 rows total

    // A operand: lanes 0-15 carry (K=0,K=1) = (x,y); lanes 16-31 carry K=2,3 = 0
    float cx = 0.f, cy = 0.f;
    if (lane < 16) {
        int g = rowBase + lane;
        int n = g >> 1;
        int pt = g & 1;
        int vx = pt ? ws[2 * NSEG + n] : ws[n];
        int vy = pt ? ws[3 * NSEG + n] : ws[NSEG + n];
        cx = 2.0f * (((float)vx + 0.5f) * (1.0f / 1024.0f)) - 1.0f;
        cy = 2.0f * (((float)vy + 0.5f) * (1.0f / 1024.0f)) - 1.0f;
    }
    v2f a; a.x = cx; a.y = cy;

    const int col = lane & 15;
    #pragma unroll
    for (int t = 0; t < 8; ++t) {
        const int c = t * 16;
        // B operand: lanes 0-15 carry rows K=0 (gauss row 0) / K=1 (gauss row 1)
        float bx = 0.f, by = 0.f;
        if (lane < 16) {
            bx = gauss[c + col];
            by = gauss[128 + c + col];
        }
        v2f b; b.x = bx; b.y = by;

        v8f acc = {};
        acc = __builtin_amdgcn_wmma_f32_16x16x4_f32(
            /*neg_a=*/false, a, /*neg_b=*/false, b,
            /*c_mod=*/(short)0, acc, /*reuse_a=*/false, /*reuse_b=*/false);

        #pragma unroll
        for (int v = 0; v < 8; ++v) {
            const int row = rowBase + ((lane < 16) ? v : v + 8);
            const int n  = row >> 1;
            const int pt = row & 1;
            const float* pc = pt ? pe3 : pe2;
            float theta = TWO_PI * acc[v];
            float s, co;
            __sincosf(theta, &s, &co);
            const int d0 = c + col;
            float* o = out + (size_t)n * 512 + pt * 256;
            o[d0]       = s  + pc[d0];
            o[d0 + 128] = co + pc[d0 + 128];
        }
    }
}

extern "C" void kernel_launch(void* const* d_in, const int* in_sizes, int n_in,
                              void* d_out, int out_size, void* d_ws, size_t ws_size,
                              hipStream_t stream) {
    // inputs: [0]=seg (unused: labels are not returned), [1]=instance_map,
    //         [2]=gauss, [3]=pe2, [4]=pe3
    const int*   imap  = (const int*)d_in[1];
    const float* gauss = (const float*)d_in[2];
    const float* pe2   = (const float*)d_in[3];
    const float* pe3   = (const float*)d_in[4];
    float*       out   = (float*)d_out;
    int*         ws    = (int*)d_ws;     // 4*NSEG ints = 64 KB

    geo_init_bbox<<<(4 * NSEG) / 256, 256, 0, stream>>>(ws);
    geo_scan<<<B_ * (H_ / 16), 256, 0, stream>>>(imap, ws);
    geo_embed<<<(2 * NSEG) / (16 * 8), 256, 0, stream>>>(ws, gauss, pe2, pe3, out);
}